// ChamferDistance_40192303956375
// MI455X (gfx1250) — compile-verified
//
#include <hip/hip_runtime.h>

// Chamfer distance on MI455X (gfx1250):
//   dists[b,n,m] = ||x_n||^2 - 2 x_n . y_m + ||y_m||^2   (C = 3)
// V_WMMA_F32_16X16X4_F32 computes the full distance tile in one op:
//   A (16x4) row n = (x0, x1, x2, 1)
//   B (4x16) col m = (-2y0, -2y1, -2y2, ||y||^2)
//   C        row n = ||x||^2  (broadcast across columns)
//   D        = full squared distance (exact f32 GEMM path)
// Row mins: per-lane min accumulate + cross-lane shuffle reduce.
// Col mins: branch-free ds_min_i32 on float bit patterns (dists >= 0),
// flushed per stage with global atomic-min.
// y-columns staged through LDS with GLOBAL_LOAD_ASYNC_TO_LDS_B128
// (ASYNCcnt) when the builtin is available.

#define B_ 4
#define C_ 3
#define N_ 8192
#define M_ 8192

#define M_STAGE 1024                 // columns staged in LDS per step
#define TILES_PER_STAGE (M_STAGE / 16)
#define NSTAGES (M_ / M_STAGE)
#define BLOCK_THREADS 256            // 8 waves
#define ROWS_PER_BLOCK 128           // 8 waves * 16 rows

typedef float v2f __attribute__((ext_vector_type(2)));
typedef float v8f __attribute__((ext_vector_type(8)));
typedef int   v4i __attribute__((ext_vector_type(4)));

typedef __attribute__((address_space(1))) v4i* gbl_v4i_ptr;
typedef __attribute__((address_space(3))) v4i* lds_v4i_ptr;

// ---- optional CDNA5 async global->LDS copy path --------------------------
#if defined(__has_builtin)
#if __has_builtin(__builtin_amdgcn_global_load_async_to_lds_b128) && \
    __has_builtin(__builtin_amdgcn_s_wait_asynccnt)
#define USE_ASYNC_COPY 1
#endif
#endif
#ifndef USE_ASYNC_COPY
#define USE_ASYNC_COPY 0
#endif

__device__ __forceinline__ void stage_copy(float4* dst, const float4* src) {
#if USE_ASYNC_COPY
  for (int i = threadIdx.x; i < M_STAGE; i += BLOCK_THREADS)
    __builtin_amdgcn_global_load_async_to_lds_b128(
        (gbl_v4i_ptr)(v4i*)(src + i), (lds_v4i_ptr)(v4i*)(dst + i), 0, 0);
#else
  for (int i = threadIdx.x; i < M_STAGE; i += BLOCK_THREADS) dst[i] = src[i];
#endif
}

__device__ __forceinline__ void stage_wait() {
#if USE_ASYNC_COPY
  __builtin_amdgcn_s_wait_asynccnt(0);
#endif
}

// ---------------------------------------------------------------- prep ----
// x2[b,n] = ||x||^2 ; y4[b,m] = {-2y0,-2y1,-2y2, ||y||^2} ; cmin = +inf bits
__global__ __launch_bounds__(256) void chamfer_prep(
    const float* __restrict__ pred, const float* __restrict__ target,
    float* __restrict__ x2, float4* __restrict__ y4, int* __restrict__ cmin) {
  const int total = B_ * N_;  // == B_ * M_
  for (int i = blockIdx.x * blockDim.x + threadIdx.x; i < total;
       i += gridDim.x * blockDim.x) {
    const int b = i / N_;
    const int n = i % N_;
    const float p0 = pred[(b * C_ + 0) * N_ + n];
    const float p1 = pred[(b * C_ + 1) * N_ + n];
    const float p2 = pred[(b * C_ + 2) * N_ + n];
    x2[i] = p0 * p0 + p1 * p1 + p2 * p2;
    const float t0 = target[(b * C_ + 0) * M_ + n];
    const float t1 = target[(b * C_ + 1) * M_ + n];
    const float t2 = target[(b * C_ + 2) * M_ + n];
    y4[i] = make_float4(-2.0f * t0, -2.0f * t1, -2.0f * t2,
                        t0 * t0 + t1 * t1 + t2 * t2);
    cmin[i] = 0x7F800000;  // +inf
  }
}

// ---------------------------------------------------------------- main ----
__global__ __launch_bounds__(BLOCK_THREADS) void chamfer_main(
    const float* __restrict__ pred, const float* __restrict__ x2,
    const float4* __restrict__ y4, float* __restrict__ rmin,
    int* __restrict__ cmin) {
  __shared__ float4 ybuf[2][M_STAGE];   // 2 x 16 KB, double buffered
  __shared__ int cbuf[M_STAGE];         // block-level column mins (int bits)

  const int lane = threadIdx.x & 31;
  const int wave = threadIdx.x >> 5;
  const int half = lane >> 4;           // 0: lanes 0-15, 1: lanes 16-31
  const int l15  = lane & 15;

  const int b    = blockIdx.x / (N_ / ROWS_PER_BLOCK);
  const int nblk = blockIdx.x % (N_ / ROWS_PER_BLOCK);
  const int n0   = nblk * ROWS_PER_BLOCK + wave * 16;

  // --- A fragment (16x4, K padded: col3 = 1.0) ---------------------------
  // Layout: lanes 0-15 -> K=0 (v0), K=1 (v1); lanes 16-31 -> K=2, K=3.
  const int row = n0 + l15;
  const float xa0 = pred[(b * C_ + 0) * N_ + row];
  const float xa1 = pred[(b * C_ + 1) * N_ + row];
  const float xa2 = pred[(b * C_ + 2) * N_ + row];
  v2f afrag;
  afrag.x = half ? xa2 : xa0;
  afrag.y = half ? 1.0f : xa1;

  // --- C fragment: ||x||^2 broadcast, matching the C/D layout ------------
  // VGPR k: lanes<16 -> row n0+k, lanes>=16 -> row n0+8+k.
  v8f cx2;
#pragma unroll
  for (int k = 0; k < 8; ++k) cx2[k] = x2[b * N_ + n0 + k + half * 8];

  v8f racc;
#pragma unroll
  for (int k = 0; k < 8; ++k) racc[k] = 3.0e38f;

  for (int i = threadIdx.x; i < M_STAGE; i += BLOCK_THREADS)
    cbuf[i] = 0x7F800000;

  const float4* ysrc = y4 + (size_t)b * M_;
  stage_copy(&ybuf[0][0], ysrc);        // preload stage 0
  stage_wait();
  __syncthreads();

  for (int s = 0; s < NSTAGES; ++s) {
    const int cur = s & 1;
    if (s + 1 < NSTAGES)                // prefetch next stage
      stage_copy(&ybuf[cur ^ 1][0], ysrc + (s + 1) * M_STAGE);

    // Per-lane B base: column l15 of each tile, half-selected float2.
    // float4 col = two adjacent float2s -> half-select folded into address.
    const float2* ylane =
        (const float2*)&ybuf[cur][0] + (l15 * 2 + half);

#pragma unroll 4
    for (int t = 0; t < TILES_PER_STAGE; ++t) {
      const float2 q = ylane[t * 32];   // tile stride = 16 float4 = 32 float2
      v2f bfrag;
      bfrag.x = q.x;                    // K=0 / K=2
      bfrag.y = q.y;                    // K=1 / K=3 (= ||y||^2 in upper half)

      // D = full squared distance tile (x2 folded in via accumulator)
      v8f d = __builtin_amdgcn_wmma_f32_16x16x4_f32(
          false, afrag, false, bfrag, (short)0, cx2, false, false);

      float cl = 3.0e38f;
#pragma unroll
      for (int k = 0; k < 8; ++k) {
        racc[k] = fminf(racc[k], d[k]);  // row-min accumulate
        cl = fminf(cl, d[k]);            // column min over the 8 rows
      }
      // Branch-free: lanes j and j+16 hit the same column; LDS atomic
      // serializes the pair. Clamp >= 0 so int ordering == float ordering.
      atomicMin(&cbuf[t * 16 + l15], __float_as_int(fmaxf(cl, 0.0f)));
    }
    stage_wait();                        // next buffer resident before flip
    __syncthreads();

    // flush block-level column mins of this stage to global, reset buffer
    const int colbase = b * M_ + s * M_STAGE;
    for (int i = threadIdx.x; i < M_STAGE; i += BLOCK_THREADS) {
      atomicMin(&cmin[colbase + i], cbuf[i]);
      cbuf[i] = 0x7F800000;
    }
    __syncthreads();
  }

  // --- row mins: reduce across the 16 lanes of each half-wave ------------
#pragma unroll
  for (int k = 0; k < 8; ++k) {
    float v = racc[k];
    v = fminf(v, __shfl_xor(v, 8, 32));
    v = fminf(v, __shfl_xor(v, 4, 32));
    v = fminf(v, __shfl_xor(v, 2, 32));
    v = fminf(v, __shfl_xor(v, 1, 32));
    if (l15 == 0) {  // lane 0 -> rows n0+k, lane 16 -> rows n0+8+k
      const int r = n0 + k + half * 8;
      rmin[b * N_ + r] = fmaxf(v, 0.0f);  // already includes ||x||^2
    }
  }
}

// ------------------------------------------------------------- finalize ---
__global__ __launch_bounds__(256) void chamfer_final(
    const float* __restrict__ rmin, const int* __restrict__ cmin,
    float* __restrict__ out) {
  __shared__ float sdata[256];
  float s = 0.0f;
  const float wr = 1.0f / (float)(B_ * N_);
  const float wc = 1.0f / (float)(B_ * M_);
  for (int i = threadIdx.x; i < B_ * N_; i += 256) s += rmin[i] * wr;
  for (int i = threadIdx.x; i < B_ * M_; i += 256)
    s += __int_as_float(cmin[i]) * wc;
  sdata[threadIdx.x] = s;
  __syncthreads();
  for (int st = 128; st > 0; st >>= 1) {
    if ((int)threadIdx.x < st) sdata[threadIdx.x] += sdata[threadIdx.x + st];
    __syncthreads();
  }
  if (threadIdx.x == 0) out[0] = sdata[0];
}

// ---------------------------------------------------------------- launch --
extern "C" void kernel_launch(void* const* d_in, const int* in_sizes, int n_in,
                              void* d_out, int out_size, void* d_ws,
                              size_t ws_size, hipStream_t stream) {
  const float* pred   = (const float*)d_in[0];  // (B,C,N) f32
  const float* target = (const float*)d_in[1];  // (B,C,M) f32

  // workspace layout (all 16B aligned):
  char* w = (char*)d_ws;
  float*  x2   = (float*)w;                                   // B*N f32
  float4* y4   = (float4*)(w + (size_t)B_ * N_ * 4);          // B*M float4
  float*  rmin = (float*)(w + (size_t)B_ * N_ * 4 +
                          (size_t)B_ * M_ * 16);              // B*N f32
  int*    cmin = (int*)(w + (size_t)B_ * N_ * 4 +
                        (size_t)B_ * M_ * 16 +
                        (size_t)B_ * N_ * 4);                 // B*M i32

  chamfer_prep<<<128, 256, 0, stream>>>(pred, target, x2, y4, cmin);

  const int nblocks = B_ * (N_ / ROWS_PER_BLOCK);  // 256
  chamfer_main<<<nblocks, BLOCK_THREADS, 0, stream>>>(pred, x2, y4, rmin, cmin);

  chamfer_final<<<1, 256, 0, stream>>>(rmin, cmin, (float*)d_out);
}